// IPAM_Module_29480655519892
// MI455X (gfx1250) — compile-verified
//
#include <hip/hip_runtime.h>

typedef __attribute__((ext_vector_type(2))) float v2f;
typedef __attribute__((ext_vector_type(8))) float v8f;

#define B_  4
#define C_  64
#define CQ_ 8
#define N_  4096
#define NT_ 256   // N/16 tiles

// ---------------- projection kernels (q,k in [b][d][n]; v transposed to [b][n][c]) ----

__global__ void qk_proj_kernel(const float* __restrict__ x,
                               const float* __restrict__ Wq, const float* __restrict__ bq,
                               const float* __restrict__ Wk, const float* __restrict__ bk,
                               float* __restrict__ qout, float* __restrict__ kout) {
  int idx = blockIdx.x * blockDim.x + threadIdx.x;   // B*CQ*N threads
  int n = idx & (N_ - 1);
  int d = (idx >> 12) & (CQ_ - 1);
  float aq = bq[d], ak = bk[d];
  const float* xb = x + (size_t)(idx >> 15) * C_ * N_ + n;
#pragma unroll 4
  for (int c = 0; c < C_; ++c) {
    float xv = xb[(size_t)c * N_];
    aq += Wq[d * C_ + c] * xv;
    ak += Wk[d * C_ + c] * xv;
  }
  qout[idx] = aq;   // idx == ((b*CQ + d)*N + n)
  kout[idx] = ak;
}

__global__ void v_proj_kernel(const float* __restrict__ x,
                              const float* __restrict__ Wv, const float* __restrict__ bv,
                              float* __restrict__ vT) {
  int idx = blockIdx.x * blockDim.x + threadIdx.x;   // B*N*C threads
  int c = idx & (C_ - 1);
  int n = (idx >> 6) & (N_ - 1);
  float a = bv[c];
  const float* xb = x + (size_t)(idx >> 18) * C_ * N_ + n;
#pragma unroll 4
  for (int cc = 0; cc < C_; ++cc)
    a += Wv[c * C_ + cc] * xb[(size_t)cc * N_];
  vT[idx] = a;      // idx == ((b*N + n)*C + c)
}

// ---------------- fused flash-attention kernel -----------------------------------------

__device__ __forceinline__ v8f wmma_f32x4(v2f a, v2f b, v8f c) {
  return __builtin_amdgcn_wmma_f32_16x16x4_f32(false, a, false, b, (short)0, c, false, false);
}

__global__ __launch_bounds__(128)
void attn_kernel(const float* __restrict__ x, const float* __restrict__ q,
                 const float* __restrict__ k, const float* __restrict__ vT,
                 const float* __restrict__ gamma, float* __restrict__ out) {
  const int b    = blockIdx.y;
  const int i0   = blockIdx.x * 16;
  const int wave = threadIdx.x >> 5;
  const int lane = threadIdx.x & 31;
  const int half = lane >> 4;
  const int l16  = lane & 15;

  const float* qb = q  + (size_t)b * CQ_ * N_;
  const float* kb = k  + (size_t)b * CQ_ * N_;
  const float* vb = vT + (size_t)b * N_ * C_;

  // Loop-invariant Q operand (B-matrix of S^T): B[kd][ni] = q[kd][i0+ni]
  v2f q_lo, q_hi;
  q_lo.x = qb[(2 * half + 0) * N_ + i0 + l16];
  q_lo.y = qb[(2 * half + 1) * N_ + i0 + l16];
  q_hi.x = qb[(4 + 2 * half + 0) * N_ + i0 + l16];
  q_hi.y = qb[(4 + 2 * half + 1) * N_ + i0 + l16];

  __shared__ float probs[2][4][16][16];   // [buf][tile-in-round][j local][i local], 8 KB
  __shared__ float smax[4][16];
  __shared__ float ssum[4][16];

  // ---- pass 1: per-row max & softmax denominator; waves split the j-range 4-way ----
  float m = -3.0e38f, s = 0.0f;
  for (int jt = wave; jt < NT_; jt += 4) {
    const int j0 = jt * 16;
    v2f a_lo, a_hi;                       // A[mj][kd] = k[kd][j0+mj]
    a_lo.x = kb[(2 * half + 0) * N_ + j0 + l16];
    a_lo.y = kb[(2 * half + 1) * N_ + j0 + l16];
    a_hi.x = kb[(4 + 2 * half + 0) * N_ + j0 + l16];
    a_hi.y = kb[(4 + 2 * half + 1) * N_ + j0 + l16];
    v8f st = {};
    st = wmma_f32x4(a_lo, q_lo, st);
    st = wmma_f32x4(a_hi, q_hi, st);
    // lane holds S^T[j0 + r + 8*half][i0 + l16], r=0..7 -> a single query row per lane
    float tmax = st[0];
#pragma unroll
    for (int r = 1; r < 8; ++r) tmax = fmaxf(tmax, st[r]);
    float nm = fmaxf(m, tmax);
    float a = 0.0f;
#pragma unroll
    for (int r = 0; r < 8; ++r) a += __expf(st[r] - nm);
    s = s * __expf(m - nm) + a;
    m = nm;
  }
  {   // combine half-groups (j 0..7 vs 8..15 of each tile)
    float pm = __shfl_xor(m, 16, 32);
    float ps = __shfl_xor(s, 16, 32);
    float nm = fmaxf(m, pm);
    s = s * __expf(m - nm) + ps * __expf(pm - nm);
    m = nm;
  }
  if (lane < 16) { smax[wave][lane] = m; ssum[wave][lane] = s; }
  __syncthreads();
  float M = -3.0e38f, S = 0.0f;
#pragma unroll
  for (int ww = 0; ww < 4; ++ww) {
    float pm = smax[ww][l16], ps = ssum[ww][l16];
    float nm = fmaxf(M, pm);
    S = S * __expf(M - nm) + ps * __expf(pm - nm);
    M = nm;
  }
  const float invS = 1.0f / S;

  // ---- pass 2: producer/consumer rounds of 4 j-tiles through LDS ----
  // Wave w produces probs for tile (4*rnd + w); all waves consume all 4 tiles for
  // their private 16-channel output slice. Double-buffered: 1 barrier per round.
  const int c0 = wave * 16;
  v8f acc = {};
  for (int rnd = 0; rnd < NT_ / 4; ++rnd) {
    const int buf = rnd & 1;
    const int j0  = (rnd * 4 + wave) * 16;
    // produce: S^T tile for this wave's j-tile
    v2f a_lo, a_hi;
    a_lo.x = kb[(2 * half + 0) * N_ + j0 + l16];
    a_lo.y = kb[(2 * half + 1) * N_ + j0 + l16];
    a_hi.x = kb[(4 + 2 * half + 0) * N_ + j0 + l16];
    a_hi.y = kb[(4 + 2 * half + 1) * N_ + j0 + l16];
    v8f st = {};
    st = wmma_f32x4(a_lo, q_lo, st);
    st = wmma_f32x4(a_hi, q_hi, st);
#pragma unroll
    for (int r = 0; r < 8; ++r) {
      float p = __expf(st[r] - M) * invS;
      p = (p > 0.5f) ? p : 0.0f;                 // threshold mask
      probs[buf][wave][r + 8 * half][l16] = p;   // conflict-free: 16 consecutive floats
    }
    __syncthreads();
    // consume: PV over the round's 64 j-values for channels [c0, c0+16)
    const int jr0 = rnd * 64;
#pragma unroll
    for (int t = 0; t < 4; ++t) {
#pragma unroll
      for (int g = 0; g < 4; ++g) {
        v2f b2;   // B2[2*half+jv][i] = attn[i][jr0 + 16t + 4g + 2*half + jv]
        b2.x = probs[buf][t][4 * g + 2 * half + 0][l16];
        b2.y = probs[buf][t][4 * g + 2 * half + 1][l16];
        // A2[mc][2*half+jv] = v[c0+mc][j...]  (vT layout -> coalesced over c)
        const float* vp = vb + (size_t)(jr0 + t * 16 + 4 * g + 2 * half) * C_ + c0 + l16;
        v2f a2;
        a2.x = vp[0];
        a2.y = vp[C_];
        acc = wmma_f32x4(a2, b2, acc);
      }
    }
  }

  // ---- epilogue: out = gamma * attn_out + x ----
  const float gm = gamma[0];
#pragma unroll
  for (int r = 0; r < 8; ++r) {
    int c = c0 + r + 8 * half;
    size_t o = ((size_t)b * C_ + c) * N_ + i0 + l16;
    out[o] = gm * acc[r] + x[o];
  }
}

// ---------------- launcher -------------------------------------------------------------

extern "C" void kernel_launch(void* const* d_in, const int* in_sizes, int n_in,
                              void* d_out, int out_size, void* d_ws, size_t ws_size,
                              hipStream_t stream) {
  const float* x     = (const float*)d_in[0];
  const float* Wq    = (const float*)d_in[1];
  const float* bq    = (const float*)d_in[2];
  const float* Wk    = (const float*)d_in[3];
  const float* bk    = (const float*)d_in[4];
  const float* Wv    = (const float*)d_in[5];
  const float* bv    = (const float*)d_in[6];
  const float* gamma = (const float*)d_in[7];
  float* out = (float*)d_out;

  float* qw = (float*)d_ws;                        // B*CQ*N
  float* kw = qw + (size_t)B_ * CQ_ * N_;          // B*CQ*N
  float* vT = kw + (size_t)B_ * CQ_ * N_;          // B*N*C

  qk_proj_kernel<<<(B_ * CQ_ * N_) / 256, 256, 0, stream>>>(x, Wq, bq, Wk, bk, qw, kw);
  v_proj_kernel<<<(B_ * N_ * C_) / 256, 256, 0, stream>>>(x, Wv, bv, vT);
  attn_kernel<<<dim3(NT_, B_), 128, 0, stream>>>(x, qw, kw, vT, gamma, out);
}